// MeanAggregator1_20538533610157
// MI455X (gfx1250) — compile-verified
//
#include <hip/hip_runtime.h>

typedef __attribute__((ext_vector_type(2))) float v2f;
typedef __attribute__((ext_vector_type(8))) float v8f;

// Hardware tanh: single V_TANH_F32 trans op on gfx1250.
__device__ __forceinline__ float fast_tanh(float x) {
#if __has_builtin(__builtin_amdgcn_tanhf)
  return __builtin_amdgcn_tanhf(x);
#else
  float y;
  // TRANS op: one independent op (v_nop) before the result may be consumed.
  asm("v_tanh_f32 %0, %1\n\tv_nop" : "=v"(y) : "v"(x));
  return y;
#endif
}

// One wave (32 lanes) per output row b.
// out[b, :] = tanh( (d_weight[b]/16) * sum_k f_weight[b,k] * embed[neigh_idx[b,k], :] )
//
// Per 16-wide d-tile: 4x V_WMMA_F32_16X16X4_F32 with
//   A[m, kk] = f_weight[b, 4c+kk]   (broadcast over m -> every D row = result)
//   B[kk, n] = embed[idx[b,4c+kk], d0+n]
// Operand layout per ISA 16x4 f32 A spec: lane-half adds +2 to K, VGPR index adds +1.
// Tiles are processed in pairs so the epilogue uses all 32 lanes:
//   lanes 0-15 read acc0 VGPR0 (row M=0), lanes 16-31 read acc1 VGPR0 (row M=8).
__global__ __launch_bounds__(256) void mean_agg_wmma_f32(
    const int*   __restrict__ neigh_idx,  // [B,16]
    const float* __restrict__ embed,      // [U,256]
    const float* __restrict__ d_weight,   // [B]
    const float* __restrict__ f_weight,   // [B,16]
    float*       __restrict__ out,        // [B,256]
    int B)
{
  const int lane = threadIdx.x & 31;
  const int wave = threadIdx.x >> 5;
  const int b    = blockIdx.x * 8 + wave;
  if (b >= B) return;                    // wave-uniform: EXEC stays all-ones

  const int half = lane >> 4;            // 0: lanes 0-15, 1: lanes 16-31
  const int n    = lane & 15;            // column within a 16-wide d tile

  const int*   idx_b = neigh_idx + (size_t)b * 16;
  const float* fw_b  = f_weight  + (size_t)b * 16;

  // Per-chunk A operands and 32-bit element offsets into embed (uniform SGPR
  // base + VGPR offset with scale_offset addressing).
  v2f a[4];
  unsigned o0[4], o1[4];
#pragma unroll
  for (int c = 0; c < 4; ++c) {
    const int kk0 = 4 * c + 2 * half;    // K index held by VGPR0 of this lane
    a[c].x = fw_b[kk0];
    a[c].y = fw_b[kk0 + 1];
    o0[c]  = (unsigned)idx_b[kk0]     * 256u;
    o1[c]  = (unsigned)idx_b[kk0 + 1] * 256u;
  }

  const float scale = d_weight[b] * (1.0f / 16.0f);
  float* out_b = out + (size_t)b * 256;

#pragma unroll
  for (int p = 0; p < 8; ++p) {          // pair of 16-wide d tiles per iter
    const unsigned d0 = (unsigned)p * 32u;
    v8f acc0 = {};
    v8f acc1 = {};
#pragma unroll
    for (int c = 0; c < 4; ++c) {
      v2f b0, b1;
      b0.x = embed[o0[c] + d0 + n];      // 16 lanes -> contiguous 64B segment
      b0.y = embed[o1[c] + d0 + n];
      b1.x = embed[o0[c] + d0 + 16 + n];
      b1.y = embed[o1[c] + d0 + 16 + n];
      // 8 args: (neg_a, A, neg_b, B, c_mod, C, reuse_a, reuse_b)
      acc0 = __builtin_amdgcn_wmma_f32_16x16x4_f32(
          false, a[c], false, b0, (short)0, acc0, false, false);
      acc1 = __builtin_amdgcn_wmma_f32_16x16x4_f32(
          false, a[c], false, b1, (short)0, acc1, false, false);
    }
    // Full-wave epilogue: lane L stores d = 32p + L (128B coalesced store).
    const float v = (half == 0) ? acc0[0] : acc1[0];
    out_b[d0 + lane] = fast_tanh(v * scale);
  }
}

extern "C" void kernel_launch(void* const* d_in, const int* in_sizes, int n_in,
                              void* d_out, int out_size, void* d_ws, size_t ws_size,
                              hipStream_t stream) {
  const int*   neigh_idx = (const int*)  d_in[0];   // [B,16] int32
  const float* embed     = (const float*)d_in[1];   // [U,256] f32
  const float* d_weight  = (const float*)d_in[2];   // [B] f32
  const float* f_weight  = (const float*)d_in[3];   // [B,16] f32
  float*       out       = (float*)d_out;           // [B,256] f32

  const int B = in_sizes[2];                        // d_weight is [B]
  const int blocks = (B + 7) / 8;                   // 8 waves per block, 1 row per wave
  mean_agg_wmma_f32<<<blocks, 256, 0, stream>>>(
      neigh_idx, embed, d_weight, f_weight, out, B);
}